// LearnableSymmetricPositionalEncoding_91156385890835
// MI455X (gfx1250) — compile-verified
//
#include <hip/hip_runtime.h>
#include <stdint.h>

// Problem constants (match the reference harness)
#define BATCH       32
#define SEQ         4096
#define DIM         1024
#define TABLE_ROWS  2049   // (SEQ-1)/2 + 2
#define ROWS_PER_BLOCK 4   // amortize per-block SALU prologue over 4 rows

typedef float v4f __attribute__((ext_vector_type(4)));
typedef unsigned int v4u __attribute__((ext_vector_type(4)));

// ---------------------------------------------------------------------------
// Kernel 1: valid_len[b] = sum(mask[b, :])  (mask = contiguous 0/1 byte prefix)
// 256 threads * 16 bytes = 4096 bytes = exactly one row.
// Bytes are 0x00/0x01, so popcount(dword) == sum of its 4 bytes.
// (Compiler lowers the LDS atomic into a v_readlane wave reduction.)
// ---------------------------------------------------------------------------
__global__ void lspe_valid_len_kernel(const uint8_t* __restrict__ mask,
                                      int* __restrict__ vl) {
    __shared__ int acc;
    if (threadIdx.x == 0) acc = 0;
    __syncthreads();

    const int b = blockIdx.x;
    const v4u* row = (const v4u*)(mask + (size_t)b * SEQ);
    v4u v = row[threadIdx.x];
    int cnt = __popc(v.x) + __popc(v.y) + __popc(v.z) + __popc(v.w);
    atomicAdd(&acc, cnt);
    __syncthreads();

    if (threadIdx.x == 0) vl[b] = acc;
}

// ---------------------------------------------------------------------------
// Kernel 2: out[b,i,:] = x[b,i,:] + (i < L[b] ? PE[clip(min(i,L-1-i)+1)] : 0)
//
// One block per (b, 4 consecutive i); 256 threads * float4 = 1024 floats/row.
// - x / out: non-temporal b128 (1 GB streaming traffic must not thrash L2)
// - PE:      regular-temporal b128 (8.4 MB table stays L2-resident, ~64x reuse)
// - branch + table index are block-uniform -> SALU control flow, no divergence
// - 4 rows share one base address; per-row access is an immediate IOFFSET
// ---------------------------------------------------------------------------
__global__ void __launch_bounds__(256)
lspe_add_kernel(const float* __restrict__ x,
                const float* __restrict__ pe,
                const int* __restrict__ vl,
                float* __restrict__ out) {
    const int i0 = blockIdx.x * ROWS_PER_BLOCK;   // first seq position (uniform)
    const int b  = blockIdx.y;                    // batch             (uniform)
    const int t  = threadIdx.x;                   // float4 chunk within a row

    const int L = vl[b];                          // uniform scalar load

    const size_t base = ((size_t)b * SEQ + (size_t)i0) * DIM;
    const v4f* xbase = (const v4f*)(x + base) + t;
    v4f*       obase = (v4f*)(out + base) + t;

#pragma unroll
    for (int r = 0; r < ROWS_PER_BLOCK; ++r) {
        const int i = i0 + r;                     // uniform
        // Streaming read of x: non-temporal so it does not evict the PE table.
        // Row r sits at immediate offset r*DIM floats from the shared base.
        v4f acc = __builtin_nontemporal_load(xbase + (size_t)r * (DIM / 4));

        if (i < L) {                              // uniform branch (s_cbranch)
            int s = min(i, L - 1 - i) + 1;        // symmetric index
            s = max(0, min(s, TABLE_ROWS - 1));
            const v4f* prow = (const v4f*)(pe + (size_t)s * DIM) + t;
            // gfx1250 prefetch path (global_prefetch_b8): warm the PE line.
            __builtin_prefetch((const void*)prow, 0, 3);
            // Regular-temporal load: keep the hot table cache-resident.
            acc += *prow;
        }

        // Streaming write of out: non-temporal (write-once, never re-read).
        __builtin_nontemporal_store(acc, obase + (size_t)r * (DIM / 4));
    }
}

// ---------------------------------------------------------------------------
// Launch
//   d_in[0] = x    (B*S*D f32)
//   d_in[1] = mask (B*S bool bytes)
//   d_in[2] = position_embedding (TABLE_ROWS*D f32)
//   d_out   = out  (B*S*D f32)
//   d_ws    = scratch: valid_len[B] ints
// ---------------------------------------------------------------------------
extern "C" void kernel_launch(void* const* d_in, const int* in_sizes, int n_in,
                              void* d_out, int out_size, void* d_ws, size_t ws_size,
                              hipStream_t stream) {
    const float*   x    = (const float*)d_in[0];
    const uint8_t* mask = (const uint8_t*)d_in[1];
    const float*   pe   = (const float*)d_in[2];
    float*         out  = (float*)d_out;
    int*           vl   = (int*)d_ws;

    lspe_valid_len_kernel<<<BATCH, 256, 0, stream>>>(mask, vl);

    dim3 grid(SEQ / ROWS_PER_BLOCK, BATCH);
    lspe_add_kernel<<<grid, 256, 0, stream>>>(x, pe, vl, out);
}